// GCNDialogueManager_85959475462429
// MI455X (gfx1250) — compile-verified
//
#include <hip/hip_runtime.h>

typedef __attribute__((ext_vector_type(2))) float v2f;
typedef __attribute__((ext_vector_type(8))) float v8f;

#define FDIM 128   // input features
#define HDIM 128   // hidden
#define ADIM 64    // actions

// ---------------------------------------------------------------- degree init
__global__ void deg_init_kernel(float* __restrict__ deg, int N) {
    int i = blockIdx.x * blockDim.x + threadIdx.x;
    if (i < N) deg[i] = 1.0f;   // self-loop contributes 1
}

// ------------------------------------------------------------- degree scatter
__global__ void deg_accum_kernel(const long long* __restrict__ dst,
                                 float* __restrict__ deg, int E) {
    int e = blockIdx.x * blockDim.x + threadIdx.x;
    if (e < E) atomicAdd(&deg[(int)dst[e]], 1.0f);
}

// ------------------------------------------------------------- dinv = rsqrt
__global__ void dinv_kernel(float* __restrict__ deg, int N) {
    int i = blockIdx.x * blockDim.x + threadIdx.x;
    if (i < N) {
        float d = deg[i];
        deg[i] = (d > 0.0f) ? rsqrtf(d) : 0.0f;  // overwrite deg with dinv
    }
}

// -------------------------------------------------- GEMM1: h = x @ W_gcn
// grid.x = N/16 blocks, 256 threads = 8 waves; wave w computes the 16x16
// output tile at columns [16w, 16w+16). K marched in steps of 4 with
// v_wmma_f32_16x16x4_f32.
__global__ __launch_bounds__(256) void gemm1_kernel(const float* __restrict__ x,
                                                    const float* __restrict__ W,
                                                    float* __restrict__ h, int N) {
    __shared__ float xs[16][FDIM];   // 8 KB tile of x
    const int block_row = blockIdx.x * 16;
    const int tid = threadIdx.x;

    // cooperative load: 16*128 floats = 512 float4, 2 per thread
    {
        const float4* src4 = (const float4*)(x + (size_t)block_row * FDIM);
        float4* dst4 = (float4*)(&xs[0][0]);
        for (int i = tid; i < 16 * FDIM / 4; i += 256) dst4[i] = src4[i];
    }
    __syncthreads();

    const int wave  = tid >> 5;        // 0..7 -> column tile
    const int lane  = tid & 31;
    const int col0  = wave * 16;
    const int mlo   = lane & 15;       // row (A) / col (B,C) within tile
    const int khalf = (lane >> 4) * 2; // lanes 0-15: K=0,1 ; lanes 16-31: K=2,3

    v8f c = {};
    for (int k0 = 0; k0 < FDIM; k0 += 4) {
        v2f a, b;
        a.x = xs[mlo][k0 + khalf];
        a.y = xs[mlo][k0 + khalf + 1];
        b.x = W[(size_t)(k0 + khalf) * HDIM + col0 + mlo];
        b.y = W[(size_t)(k0 + khalf + 1) * HDIM + col0 + mlo];
        c = __builtin_amdgcn_wmma_f32_16x16x4_f32(false, a, false, b,
                                                  (short)0, c, false, false);
    }

    // C/D layout: VGPR r -> row r (lanes 0-15) or row 8+r (lanes 16-31)
    const int rbase = (lane >> 4) * 8;
    float* out = h + (size_t)(block_row + rbase) * HDIM + col0 + mlo;
#pragma unroll
    for (int r = 0; r < 8; ++r) out[(size_t)r * HDIM] = c[r];
}

// ------------------------------- agg = b_gcn + h * dinv^2   (self-loop term)
__global__ void agg_init_kernel(const float* __restrict__ h,
                                const float* __restrict__ dinv,
                                const float* __restrict__ b_gcn,
                                float* __restrict__ agg, int N) {
    int i = blockIdx.x * blockDim.x + threadIdx.x;
    int total = N * HDIM;
    if (i < total) {
        int node = i >> 7;           // /128
        int f    = i & (HDIM - 1);
        float di = dinv[node];
        agg[i] = b_gcn[f] + h[i] * (di * di);
    }
}

// --------------------------------------------- edge scatter (L2 atomics)
// one wave32 per edge: wave-uniform indices/norm, float4 gather per lane,
// 4 per-lane global_atomic_add_f32 -> 512B fully-coalesced per edge.
__global__ __launch_bounds__(256) void edge_scatter_kernel(
        const float* __restrict__ h,
        const long long* __restrict__ src,
        const long long* __restrict__ dst,
        const float* __restrict__ dinv,
        float* __restrict__ agg, int E) {
    int wave = (blockIdx.x * blockDim.x + threadIdx.x) >> 5;
    int lane = threadIdx.x & 31;
    if (wave >= E) return;
    int s = (int)src[wave];
    int d = (int)dst[wave];
    float nrm = dinv[s] * dinv[d];
    float4 v = ((const float4*)(h + (size_t)s * HDIM))[lane];
    float* a = agg + (size_t)d * HDIM + lane * 4;
    atomicAdd(a + 0, v.x * nrm);
    atomicAdd(a + 1, v.y * nrm);
    atomicAdd(a + 2, v.z * nrm);
    atomicAdd(a + 3, v.w * nrm);
}

// ---------------------------------------- GEMM2: out = agg @ W_act + b_act
// grid.x = N/16, 128 threads = 4 waves; wave w -> cols [16w,16w+16) of 64.
__global__ __launch_bounds__(128) void gemm2_kernel(const float* __restrict__ agg,
                                                    const float* __restrict__ Wa,
                                                    const float* __restrict__ ba,
                                                    float* __restrict__ out, int N) {
    __shared__ float as[16][HDIM];   // 8 KB tile of agg
    const int block_row = blockIdx.x * 16;
    const int tid = threadIdx.x;

    {
        const float4* src4 = (const float4*)(agg + (size_t)block_row * HDIM);
        float4* dst4 = (float4*)(&as[0][0]);
        for (int i = tid; i < 16 * HDIM / 4; i += 128) dst4[i] = src4[i];
    }
    __syncthreads();

    const int wave  = tid >> 5;        // 0..3
    const int lane  = tid & 31;
    const int col0  = wave * 16;
    const int mlo   = lane & 15;
    const int khalf = (lane >> 4) * 2;

    // preload bias into the accumulator fragment: every VGPR r has the same
    // column (lane) -> same bias value.
    float bias = ba[col0 + mlo];
    v8f c;
#pragma unroll
    for (int r = 0; r < 8; ++r) c[r] = bias;

    for (int k0 = 0; k0 < HDIM; k0 += 4) {
        v2f a, b;
        a.x = as[mlo][k0 + khalf];
        a.y = as[mlo][k0 + khalf + 1];
        b.x = Wa[(size_t)(k0 + khalf) * ADIM + col0 + mlo];
        b.y = Wa[(size_t)(k0 + khalf + 1) * ADIM + col0 + mlo];
        c = __builtin_amdgcn_wmma_f32_16x16x4_f32(false, a, false, b,
                                                  (short)0, c, false, false);
    }

    const int rbase = (lane >> 4) * 8;
    float* o = out + (size_t)(block_row + rbase) * ADIM + col0 + mlo;
#pragma unroll
    for (int r = 0; r < 8; ++r) o[(size_t)r * ADIM] = c[r];
}

// ================================================================= launcher
extern "C" void kernel_launch(void* const* d_in, const int* in_sizes, int n_in,
                              void* d_out, int out_size, void* d_ws, size_t ws_size,
                              hipStream_t stream) {
    const float*     x      = (const float*)d_in[0];
    const long long* eidx   = (const long long*)d_in[1];   // int64 [2, E]
    const float*     W_gcn  = (const float*)d_in[2];
    const float*     b_gcn  = (const float*)d_in[3];
    const float*     W_act  = (const float*)d_in[4];
    const float*     b_act  = (const float*)d_in[5];
    float*           out    = (float*)d_out;

    const int N = in_sizes[0] / FDIM;     // 50000
    const int E = in_sizes[1] / 2;        // 600000
    const long long* src = eidx;          // row 0
    const long long* dst = eidx + E;      // row 1

    // workspace layout (floats): [dinv: pad64(N)] [h: N*H] [agg: N*H]
    float* ws   = (float*)d_ws;
    const size_t npad = ((size_t)N + 63) & ~(size_t)63;
    float* dinv = ws;                      // holds deg, then dinv in-place
    float* h    = ws + npad;
    float* agg  = h + (size_t)N * HDIM;

    // 1) deg = 1 (self loops)
    deg_init_kernel<<<(N + 255) / 256, 256, 0, stream>>>(dinv, N);
    // 2) deg += 1 per incoming edge
    deg_accum_kernel<<<(E + 255) / 256, 256, 0, stream>>>(dst, dinv, E);
    // 3) dinv = rsqrt(deg)
    dinv_kernel<<<(N + 255) / 256, 256, 0, stream>>>(dinv, N);
    // 4) h = x @ W_gcn   (N is a multiple of 16: 50000 = 3125*16)
    gemm1_kernel<<<N / 16, 256, 0, stream>>>(x, W_gcn, h, N);
    // 5) agg = b_gcn + h*dinv^2  (self-loop message folded in)
    agg_init_kernel<<<(N * HDIM + 255) / 256, 256, 0, stream>>>(h, dinv, b_gcn, agg, N);
    // 6) scatter edge messages with f32 atomics (one wave per edge)
    edge_scatter_kernel<<<(E * 32 + 255) / 256, 256, 0, stream>>>(h, src, dst, dinv, agg, E);
    // 7) out = agg @ W_act + b_act
    gemm2_kernel<<<N / 16, 128, 0, stream>>>(agg, W_act, b_act, out, N);
}